// InteractionModule_50483045597845
// MI455X (gfx1250) — compile-verified
//
#include <hip/hip_runtime.h>
#include <hip/hip_bf16.h>
#include <float.h>
#include <math.h>

typedef __attribute__((ext_vector_type(2))) float v2f;
typedef __attribute__((ext_vector_type(8))) float v8f;

#define B_GR   256
#define M_AT   512
#define KNN    33
#define RADIUS 5.0f
#define LSTR   516   // padded LDS stride (floats) for the d2 strip

// One block handles one (graph, 16-row strip): grid = 256 * 32 = 8192 blocks.
// 512 threads = 16 wave32 waves.
//   Phase 1: stage 512 positions (float4 xyz0) + squared norms into LDS.
//   Phase 2: V_WMMA_F32_16X16X4_F32 Gram tiles -> d2 strip (16 x 512) in LDS.
//   Phase 3: one wave per row, 33x iterative wave-wide argmin extraction.
__global__ __launch_bounds__(512)
void knn_edges_wmma_kernel(const float* __restrict__ pos,
                           float* __restrict__ out) {
    __shared__ float4 spos[M_AT];          // 8 KB
    __shared__ float  ssq[M_AT];           // 2 KB
    __shared__ float  sd2[16 * LSTR];      // ~32 KB

    const int tid   = threadIdx.x;
    const int lane  = tid & 31;
    const int wave  = tid >> 5;
    const int blk   = blockIdx.x;
    const int b     = blk >> 5;            // graph id
    const int strip = blk & 31;            // which 16-row strip
    const int i0    = strip << 4;

    // ---- Phase 1: stage graph positions + squared norms ----
    {
        const int g = b * M_AT + tid;
        const float x = pos[3 * g + 0];
        const float y = pos[3 * g + 1];
        const float z = pos[3 * g + 2];
        spos[tid] = make_float4(x, y, z, 0.0f);
        ssq[tid]  = x * x + y * y + z * z;
    }
    __syncthreads();

    // ---- Phase 2: WMMA 16x16x4 f32 Gram tiles over the strip ----
    {
        // A fragment (16x4 f32): lane L -> row M = L&15; VGPR pair holds
        // K = {0,1} for lanes 0-15 and K = {2,3} for lanes 16-31.
        const int   ar = i0 + (lane & 15);
        const int   kb = (lane >> 4) << 1;          // 0 or 2
        const float* pa = (const float*)&spos[ar];
        v2f a;
        a.x = pa[kb];
        a.y = pa[kb + 1];

        const int mrow = (lane >> 4) << 3;          // D-tile row base: 0 or 8
        const int ncol = lane & 15;                 // D-tile column

        #pragma unroll
        for (int tt = 0; tt < 2; ++tt) {
            const int t  = wave + tt * 16;          // j-tile index (0..31)
            const int j0 = t << 4;

            // B fragment (4x16 f32): lane L -> col N = L&15, same K pairing.
            const float* pb = (const float*)&spos[j0 + (lane & 15)];
            v2f bb;
            bb.x = pb[kb];
            bb.y = pb[kb + 1];

            v8f c = {};
            c = __builtin_amdgcn_wmma_f32_16x16x4_f32(
                    false, a, false, bb, (short)0, c, false, false);

            const float sqj = ssq[j0 + ncol];
            #pragma unroll
            for (int v = 0; v < 8; ++v) {
                const int   m  = v + mrow;
                float d2 = ssq[i0 + m] + sqj - 2.0f * c[v];
                d2 = fmaxf(d2, 0.0f);
                sd2[m * LSTR + j0 + ncol] = d2;
            }
        }
    }
    __syncthreads();

    // ---- Phase 3: per-row top-33 extraction (wave `w` owns strip row `w`) ----
    {
        const int m = wave;                         // 0..15
        float cd[16];
        #pragma unroll
        for (int ci = 0; ci < 16; ++ci)
            cd[ci] = sd2[m * LSTR + lane + (ci << 5)];   // candidate j = lane + 32*ci

        const int    dsti = b * M_AT + i0 + m;
        const float4 pd   = spos[i0 + m];
        const int    E    = B_GR * M_AT * KNN;

        for (int k = 0; k < KNN; ++k) {
            // lane-local argmin; ascending ci => ascending j, strict '<'
            // keeps the smallest index on ties (matches top_k tie-break)
            float bd = FLT_MAX;
            int   bi = -1;
            #pragma unroll
            for (int ci = 0; ci < 16; ++ci) {
                if (cd[ci] < bd) { bd = cd[ci]; bi = lane + (ci << 5); }
            }
            // wave32 butterfly reduce on (d2, idx) with tie-break on idx
            #pragma unroll
            for (int off = 16; off > 0; off >>= 1) {
                const float od = __shfl_xor(bd, off, 32);
                const int   oi = __shfl_xor(bi, off, 32);
                if (od < bd ||
                    (od == bd && (unsigned)oi < (unsigned)bi)) {
                    bd = od; bi = oi;
                }
            }
            // emit edge k for this dst (closest-first ordering, self at k=0)
            if (lane == 0) {
                const int    e    = dsti * KNN + k;
                const int    srci = b * M_AT + bi;
                const float4 ps   = spos[bi];
                const float dx = ps.x - pd.x;
                const float dy = ps.y - pd.y;
                const float dz = ps.z - pd.z;
                const float dd = dx * dx + dy * dy + dz * dz;
                const float dist = (dd > 0.0f) ? sqrtf(dd) : 0.0f;
                out[e]         = (float)srci;                    // edge_index[0]
                out[E + e]     = (float)dsti;                    // edge_index[1]
                out[2 * E + e] = dist;                           // dist
                out[3 * E + e] = (dist <= RADIUS) ? 1.0f : 0.0f; // mask
            }
            // invalidate the extracted candidate in its owning lane
            if (bi >= 0 && (bi & 31) == lane) cd[bi >> 5] = FLT_MAX;
        }
    }
}

extern "C" void kernel_launch(void* const* d_in, const int* in_sizes, int n_in,
                              void* d_out, int out_size, void* d_ws, size_t ws_size,
                              hipStream_t stream) {
    const float* pos = (const float*)d_in[0];   // [B*M, 3] float32
    // d_in[1] = batch (int32) unused: uniform graph size M => id // M
    float* out = (float*)d_out;                 // [4*E] floats

    dim3 grid(B_GR * (M_AT / 16));              // 8192 blocks
    dim3 block(512);                            // 16 waves (wave32)
    knn_edges_wmma_kernel<<<grid, block, 0, stream>>>(pos, out);
}